// GruAttCosMeanNet_39591008535146
// MI455X (gfx1250) — compile-verified
//
#include <hip/hip_runtime.h>
#include <math.h>

typedef float v2f __attribute__((ext_vector_type(2)));
typedef float v8f __attribute__((ext_vector_type(8)));

#define HSZ 256   // hidden size
#define H3  768   // 3*H
#define BB  16    // batch

__device__ __forceinline__ v8f wmma_f32_16x16x4(v2f a, v2f b, v8f c) {
  // D = A(16x4 f32) * B(4x16 f32) + C(16x16 f32)
  return __builtin_amdgcn_wmma_f32_16x16x4_f32(false, a, false, b, (short)0, c,
                                               false, false);
}

// Generic fp32 WMMA GEMM:
// out[batch*sO_batch + t*sO_t + m*N + (n0+n)] =
//   sum_k A[batch*sA_batch + tmap(t)*sA_t + m*sA_b + k] * W[(n0+n)*ldW + k] + bias[n0+n]
// Rows per t are exactly B=16 (one WMMA M-tile). One wave per 16x16 tile.
__global__ __launch_bounds__(32) void k_gemm(
    const float* __restrict__ A, const float* __restrict__ W,
    const float* __restrict__ bias, float* __restrict__ out,
    int K, int N, int T, int rev,
    long sA_t, long sA_b, long sA_batch, long sO_t, long sO_batch, int ldW)
{
  int n0    = blockIdx.x * 16;
  int t     = blockIdx.y;
  int batch = blockIdx.z;
  int tm    = rev ? (T - 1 - t) : t;
  int lane  = threadIdx.x;
  int mn    = lane & 15;          // A row (m) == B col (n) for this lane
  int kp    = (lane >> 4) << 1;   // k sub-offset {0,2}

  const float* Arow = A + batch * sA_batch + (long)tm * sA_t + (long)mn * sA_b;
  const float* Wrow = W + (long)(n0 + mn) * ldW;

  v8f acc = {0.f,0.f,0.f,0.f,0.f,0.f,0.f,0.f};
  for (int k0 = 0; k0 < K; k0 += 4) {
    v2f a, b;
    a[0] = Arow[k0 + kp];     a[1] = Arow[k0 + kp + 1];
    b[0] = Wrow[k0 + kp];     b[1] = Wrow[k0 + kp + 1];
    acc = wmma_f32_16x16x4(a, b, acc);
  }
  int   half = lane >> 4;
  float bv   = bias ? bias[n0 + mn] : 0.f;
  float* orow = out + batch * sO_batch + (long)t * sO_t;
#pragma unroll
  for (int i = 0; i < 8; ++i)
    orow[(long)(i + half * 8) * N + n0 + mn] = acc[i] + bv;
}

// Persistent GRU scan. One block = one (sequence, direction).
// grid.x = 2*nbatch: blockIdx.x&1 -> direction, >>1 -> batch.
// xp_* are precomputed input projections in scan order [T,16,768].
// out is [T,16,512]; fwd writes cols 0..255 at t, bwd writes cols 256..511 at T-1-t.
__global__ __launch_bounds__(512) void k_gru_scan(
    const float* __restrict__ xp_f, const float* __restrict__ xp_b,
    const float* __restrict__ Wh_f, const float* __restrict__ bh_f,
    const float* __restrict__ Wh_b, const float* __restrict__ bh_b,
    float* __restrict__ out, int T, long sXpBatch, long sOutBatch)
{
  int rev   = blockIdx.x & 1;
  int batch = blockIdx.x >> 1;
  const float* xp = (rev ? xp_b : xp_f) + (long)batch * sXpBatch;
  const float* Wh = rev ? Wh_b : Wh_f;
  const float* bh = rev ? bh_b : bh_f;
  float* outp = out + (long)batch * sOutBatch;

  __shared__ float h_lds[16][HSZ + 4];   // padded: row stride 260 -> no bank conflicts
  __shared__ float hp_lds[16][H3 + 4];

  int tid  = threadIdx.x;
  int lane = tid & 31;
  int wave = tid >> 5;            // 0..15
  int mn   = lane & 15;
  int kp   = (lane >> 4) << 1;
  int half = lane >> 4;

  for (int i = tid; i < 16 * HSZ; i += 512) h_lds[i >> 8][i & 255] = 0.f;
  // Warm Wh into WGP$/L2 (global_prefetch_b8)
  for (int r = tid; r < H3; r += 512)
    __builtin_prefetch(Wh + (long)r * HSZ, 0, 0);
  __syncthreads();

  for (int t = 0; t < T; ++t) {
    // --- hp = h @ Wh^T + bh  (M=16, N=768, K=256), 3 N-tiles per wave ---
    for (int tile = 0; tile < 3; ++tile) {
      int n0 = (wave + tile * 16) * 16;
      const float* Wrow = Wh + (long)(n0 + mn) * HSZ;
      v8f acc = {0.f,0.f,0.f,0.f,0.f,0.f,0.f,0.f};
      for (int k0 = 0; k0 < HSZ; k0 += 4) {
        v2f a, b;
        a[0] = h_lds[mn][k0 + kp];  a[1] = h_lds[mn][k0 + kp + 1];
        b[0] = Wrow[k0 + kp];       b[1] = Wrow[k0 + kp + 1];
        acc = wmma_f32_16x16x4(a, b, acc);
      }
      float bv = bh[n0 + mn];
#pragma unroll
      for (int i = 0; i < 8; ++i)
        hp_lds[i + half * 8][n0 + mn] = acc[i] + bv;
    }
    __syncthreads();

    // --- gates: r,z = sigmoid, n = tanh, h' = (1-z)*n + z*h ---
    const float* xpt = xp + (long)t * BB * H3;
    int tout = rev ? (T - 1 - t) : t;
    int cb   = rev ? HSZ : 0;
#pragma unroll
    for (int e = 0; e < 8; ++e) {
      int flat = tid + e * 512;            // 0..4095
      int b = flat >> 8, j = flat & 255;
      float hprev = h_lds[b][j];
      float xr = xpt[b * H3 + j];
      float xz = xpt[b * H3 + 256 + j];
      float xn = xpt[b * H3 + 512 + j];
      float r  = 1.f / (1.f + __expf(-(xr + hp_lds[b][j])));
      float z  = 1.f / (1.f + __expf(-(xz + hp_lds[b][256 + j])));
      float n  = tanhf(xn + r * hp_lds[b][512 + j]);
      float hn = (1.f - z) * n + z * hprev;
      h_lds[b][j] = hn;
      outp[((long)tout * BB + b) * 512 + cb + j] = hn;
    }
    __syncthreads();
  }
}

// e[b,q,c] = sum_h v[h] * tanh(opt_q[q,b,h] + ctx_key[c,b,h]); one wave per dot.
__global__ __launch_bounds__(32) void k_energy(
    const float* __restrict__ opt_q, const float* __restrict__ ctx_key,
    const float* __restrict__ v, float* __restrict__ E)
{
  int c = blockIdx.x, q = blockIdx.y, b = blockIdx.z;
  int lane = threadIdx.x;
  const float* oq = opt_q   + ((long)q * BB + b) * HSZ;
  const float* ck = ctx_key + ((long)c * BB + b) * HSZ;
  float s = 0.f;
  for (int h = lane; h < HSZ; h += 32) s += v[h] * tanhf(oq[h] + ck[h]);
  for (int off = 16; off; off >>= 1) s += __shfl_down(s, off, 32);
  if (lane == 0) E[((long)b * 64 + q) * 128 + c] = s;
}

// softmax over q (64) of E[b,:,c], then a_ctx[c,b,h] = sum_q w_q * opt_q[q,b,h]
__global__ __launch_bounds__(256) void k_actx(
    const float* __restrict__ E, const float* __restrict__ opt_q,
    float* __restrict__ a_ctx)
{
  int c = blockIdx.x, b = blockIdx.y, tid = threadIdx.x;
  __shared__ float w[64];
  __shared__ float s_inv;
  if (tid < 64) w[tid] = E[((long)b * 64 + tid) * 128 + c];
  __syncthreads();
  if (tid == 0) {
    float mx = -1e30f;
    for (int q = 0; q < 64; ++q) mx = fmaxf(mx, w[q]);
    float sm = 0.f;
    for (int q = 0; q < 64; ++q) { w[q] = __expf(w[q] - mx); sm += w[q]; }
    s_inv = 1.f / sm;
  }
  __syncthreads();
  float inv = s_inv, acc = 0.f;
  for (int q = 0; q < 64; ++q) acc += w[q] * opt_q[((long)q * BB + b) * HSZ + tid];
  a_ctx[((long)c * BB + b) * HSZ + tid] = acc * inv;
}

// softmax over c (128) of E[b,q,:], then a_opt[q,b,h] = sum_c w_c * ctx_key[c,b,h]
__global__ __launch_bounds__(256) void k_aopt(
    const float* __restrict__ E, const float* __restrict__ ctx_key,
    float* __restrict__ a_opt)
{
  int q = blockIdx.x, b = blockIdx.y, tid = threadIdx.x;
  __shared__ float w[128];
  __shared__ float s_inv;
  if (tid < 128) w[tid] = E[((long)b * 64 + q) * 128 + tid];
  __syncthreads();
  if (tid == 0) {
    float mx = -1e30f;
    for (int c = 0; c < 128; ++c) mx = fmaxf(mx, w[c]);
    float sm = 0.f;
    for (int c = 0; c < 128; ++c) { w[c] = __expf(w[c] - mx); sm += w[c]; }
    s_inv = 1.f / sm;
  }
  __syncthreads();
  float inv = s_inv, acc = 0.f;
  for (int c = 0; c < 128; ++c) acc += w[c] * ctx_key[((long)c * BB + b) * HSZ + tid];
  a_opt[((long)q * BB + b) * HSZ + tid] = acc * inv;
}

// m[b, h2] = mean_t in[t,b,h2], h2 in [0,512)
__global__ __launch_bounds__(512) void k_mean(
    const float* __restrict__ in, float* __restrict__ m, int T)
{
  int b = blockIdx.x, h2 = threadIdx.x;
  float s = 0.f;
  for (int t = 0; t < T; ++t) s += in[((long)t * BB + b) * 512 + h2];
  m[b * 512 + h2] = s / (float)T;
}

// logits[b*5+opt] = cos(m_ctx[b], m_opt[b]) with eps-clamped norms
__global__ __launch_bounds__(512) void k_cosine(
    const float* __restrict__ mc, const float* __restrict__ mo,
    float* __restrict__ logits, int opt)
{
  int b = blockIdx.x, tid = threadIdx.x;
  __shared__ float sd[512], sa[512], sb[512];
  float x = mc[b * 512 + tid], y = mo[b * 512 + tid];
  sd[tid] = x * y; sa[tid] = x * x; sb[tid] = y * y;
  __syncthreads();
  for (int off = 256; off; off >>= 1) {
    if (tid < off) { sd[tid] += sd[tid+off]; sa[tid] += sa[tid+off]; sb[tid] += sb[tid+off]; }
    __syncthreads();
  }
  if (tid == 0) {
    float na = fmaxf(sqrtf(sa[0]), 1e-8f);
    float nb = fmaxf(sqrtf(sb[0]), 1e-8f);
    logits[b * 5 + opt] = sd[0] / (na * nb);
  }
}

// out[b, opt] = softmax over 5 options
__global__ __launch_bounds__(32) void k_final(
    const float* __restrict__ logits, float* __restrict__ out)
{
  int b = threadIdx.x;
  if (b >= 16) return;
  float e[5], mx = -1e30f, sm = 0.f;
  for (int o = 0; o < 5; ++o) mx = fmaxf(mx, logits[b * 5 + o]);
  for (int o = 0; o < 5; ++o) { e[o] = __expf(logits[b * 5 + o] - mx); sm += e[o]; }
  for (int o = 0; o < 5; ++o) out[b * 5 + o] = e[o] / sm;
}

extern "C" void kernel_launch(void* const* d_in, const int* in_sizes, int n_in,
                              void* d_out, int out_size, void* d_ws, size_t ws_size,
                              hipStream_t stream)
{
  (void)in_sizes; (void)n_in; (void)out_size; (void)ws_size;
  const float* context  = (const float*)d_in[0];
  const float* options  = (const float*)d_in[2];
  const float* rnn_Wi_f = (const float*)d_in[4];
  const float* rnn_Wh_f = (const float*)d_in[5];
  const float* rnn_bi_f = (const float*)d_in[6];
  const float* rnn_bh_f = (const float*)d_in[7];
  const float* rnn_Wi_b = (const float*)d_in[8];
  const float* rnn_Wh_b = (const float*)d_in[9];
  const float* rnn_bi_b = (const float*)d_in[10];
  const float* rnn_bh_b = (const float*)d_in[11];
  const float* att_Wi_f = (const float*)d_in[12];
  const float* att_Wh_f = (const float*)d_in[13];
  const float* att_bi_f = (const float*)d_in[14];
  const float* att_bh_f = (const float*)d_in[15];
  const float* att_Wi_b = (const float*)d_in[16];
  const float* att_Wh_b = (const float*)d_in[17];
  const float* att_bi_b = (const float*)d_in[18];
  const float* att_bh_b = (const float*)d_in[19];
  const float* Wk       = (const float*)d_in[20];
  const float* Wq       = (const float*)d_in[21];
  const float* v_energy = (const float*)d_in[22];

  const int LC = 128, LO = 64, NOPT = 5, E = 300;

  float* ws = (float*)d_ws;
  size_t off = 0;
  auto alloc = [&](size_t n) { float* p = ws + off; off += n; return p; };

  float* xp_big_f = alloc((size_t)LC * 16 * 768);       // ctx proj fwd; reused as att a_ctx proj fwd
  float* xp_big_b = alloc((size_t)LC * 16 * 768);       // ctx proj bwd; reused as att a_ctx proj bwd
  float* gru_ctx  = alloc((size_t)LC * 16 * 512);
  float* ctx_key  = alloc((size_t)LC * 16 * 256);
  float* xp_opt_f = alloc((size_t)NOPT * LO * 16 * 768); // opt proj fwd; head reused as att a_opt proj fwd
  float* xp_opt_b = alloc((size_t)NOPT * LO * 16 * 768);
  float* gru_opt  = alloc((size_t)NOPT * LO * 16 * 512);
  float* opt_q    = alloc((size_t)NOPT * LO * 16 * 256);
  float* Ebuf     = alloc((size_t)16 * 64 * 128);
  float* a_ctx    = alloc((size_t)LC * 16 * 256);
  float* a_opt    = alloc((size_t)LO * 16 * 256);
  float* att_ctx_out = alloc((size_t)LC * 16 * 512);
  float* att_opt_out = alloc((size_t)LO * 16 * 512);
  float* m_ctx    = alloc(16 * 512);
  float* m_opt    = alloc(16 * 512);
  float* logits   = alloc(16 * 5);

  // ---- context GRU ----
  k_gemm<<<dim3(48, LC, 1), 32, 0, stream>>>(context, rnn_Wi_f, rnn_bi_f, xp_big_f,
      E, 768, LC, 0, (long)E, (long)LC * E, 0L, 16L * 768, 0L, E);
  k_gemm<<<dim3(48, LC, 1), 32, 0, stream>>>(context, rnn_Wi_b, rnn_bi_b, xp_big_b,
      E, 768, LC, 1, (long)E, (long)LC * E, 0L, 16L * 768, 0L, E);
  k_gru_scan<<<2, 512, 0, stream>>>(xp_big_f, xp_big_b,
      rnn_Wh_f, rnn_bh_f, rnn_Wh_b, rnn_bh_b, gru_ctx, LC, 0L, 0L);
  k_gemm<<<dim3(16, LC, 1), 32, 0, stream>>>(gru_ctx, Wk, nullptr, ctx_key,
      512, 256, LC, 0, 16L * 512, 512L, 0L, 16L * 256, 0L, 512);

  // ---- option GRUs (all 5, batched) ----
  k_gemm<<<dim3(48, LO, NOPT), 32, 0, stream>>>(options, rnn_Wi_f, rnn_bi_f, xp_opt_f,
      E, 768, LO, 0, (long)E, (long)NOPT * LO * E, (long)LO * E, 16L * 768, (long)LO * 16 * 768, E);
  k_gemm<<<dim3(48, LO, NOPT), 32, 0, stream>>>(options, rnn_Wi_b, rnn_bi_b, xp_opt_b,
      E, 768, LO, 1, (long)E, (long)NOPT * LO * E, (long)LO * E, 16L * 768, (long)LO * 16 * 768, E);
  k_gru_scan<<<2 * NOPT, 512, 0, stream>>>(xp_opt_f, xp_opt_b,
      rnn_Wh_f, rnn_bh_f, rnn_Wh_b, rnn_bh_b, gru_opt, LO,
      (long)LO * 16 * 768, (long)LO * 16 * 512);
  k_gemm<<<dim3(16, LO, NOPT), 32, 0, stream>>>(gru_opt, Wq, nullptr, opt_q,
      512, 256, LO, 0, 16L * 512, 512L, (long)LO * 16 * 512, 16L * 256, (long)LO * 16 * 256, 512);

  // buffers free after the scans above -> alias for attention-GRU input projections
  float* xp_ac_f = xp_big_f;  float* xp_ac_b = xp_big_b;   // [128,16,768]
  float* xp_ao_f = xp_opt_f;  float* xp_ao_b = xp_opt_b;   // first [64,16,768] of each

  for (int opt = 0; opt < NOPT; ++opt) {
    const float* oq = opt_q + (size_t)opt * LO * 16 * 256;

    k_energy<<<dim3(128, 64, 16), 32, 0, stream>>>(oq, ctx_key, v_energy, Ebuf);
    k_actx<<<dim3(128, 16), 256, 0, stream>>>(Ebuf, oq, a_ctx);
    k_aopt<<<dim3(64, 16), 256, 0, stream>>>(Ebuf, ctx_key, a_opt);

    // attention-GRU over a_ctx (T=128, input dim 256)
    k_gemm<<<dim3(48, LC, 1), 32, 0, stream>>>(a_ctx, att_Wi_f, att_bi_f, xp_ac_f,
        256, 768, LC, 0, 16L * 256, 256L, 0L, 16L * 768, 0L, 256);
    k_gemm<<<dim3(48, LC, 1), 32, 0, stream>>>(a_ctx, att_Wi_b, att_bi_b, xp_ac_b,
        256, 768, LC, 1, 16L * 256, 256L, 0L, 16L * 768, 0L, 256);
    k_gru_scan<<<2, 512, 0, stream>>>(xp_ac_f, xp_ac_b,
        att_Wh_f, att_bh_f, att_Wh_b, att_bh_b, att_ctx_out, LC, 0L, 0L);

    // attention-GRU over a_opt (T=64)
    k_gemm<<<dim3(48, LO, 1), 32, 0, stream>>>(a_opt, att_Wi_f, att_bi_f, xp_ao_f,
        256, 768, LO, 0, 16L * 256, 256L, 0L, 16L * 768, 0L, 256);
    k_gemm<<<dim3(48, LO, 1), 32, 0, stream>>>(a_opt, att_Wi_b, att_bi_b, xp_ao_b,
        256, 768, LO, 1, 16L * 256, 256L, 0L, 16L * 768, 0L, 256);
    k_gru_scan<<<2, 512, 0, stream>>>(xp_ao_f, xp_ao_b,
        att_Wh_f, att_bh_f, att_Wh_b, att_bh_b, att_opt_out, LO, 0L, 0L);

    k_mean<<<16, 512, 0, stream>>>(att_ctx_out, m_ctx, LC);
    k_mean<<<16, 512, 0, stream>>>(att_opt_out, m_opt, LO);
    k_cosine<<<16, 512, 0, stream>>>(m_ctx, m_opt, logits, opt);
  }

  k_final<<<1, 32, 0, stream>>>(logits, (float*)d_out);
}